// CenterNetDecoder_19267223290026
// MI455X (gfx1250) — compile-verified
//
#include <hip/hip_runtime.h>
#include <hip/hip_bf16.h>
#include <stdint.h>

// CenterNet decode for MI455X (gfx1250, wave32).
// Memory-bound (~92 MB in -> ~4us floor @ 23.3 TB/s). No GEMM -> no WMMA;
// the CDNA5-specific path used is the async global->LDS data mover
// (global_load_async_to_lds_b128 / s_wait_asynccnt, ASYNCcnt) + prefetch.
// Stage 1 keeps LDS at 64KB/block (keep-mask + in-place rescore) so ~4
// workgroups fit per 320KB WGP, maximizing in-flight async copies.

#define NB   16
#define NC   80
#define NH   128
#define NW_  128
#define HWT  (NH * NW_)     // 16384 pixels per channel
#define KTOP 100
#define CANDP 8192          // per-batch padded candidate slots (80*100 = 8000 used)
#define NT   256            // threads per block = 8 waves (wave32)
#define NWAVES 8

__device__ __forceinline__ uint32_t lds_byte_addr(const void* p) {
  // Generic pointer to LDS: flat addr = {aperture_hi, lds_offset[31:0]}
  return (uint32_t)(uintptr_t)p;
}

__device__ __forceinline__ void async_g2l_b128(uint32_t ldsb, const void* g) {
  // ASYNCcnt-tracked DMA of 16B/lane from global into LDS (CDNA5).
  asm volatile("global_load_async_to_lds_b128 %0, %1, off"
               :: "v"(ldsb), "v"((unsigned long long)(uintptr_t)g)
               : "memory");
}

__device__ __forceinline__ void async_g2l_b128_nt(uint32_t ldsb, const void* g) {
  // Non-temporal variant for stream-once data (heatmap read exactly once).
  asm volatile("global_load_async_to_lds_b128 %0, %1, off th:TH_LOAD_NT"
               :: "v"(ldsb), "v"((unsigned long long)(uintptr_t)g)
               : "memory");
}

__device__ __forceinline__ void wait_async_zero() {
  asm volatile("s_wait_asynccnt 0x0" ::: "memory");
}

__device__ __forceinline__ void argmax_pair(float& v, int& i, float v2, int i2) {
  if (v2 > v || (v2 == v && i2 < i)) { v = v2; i = i2; }
}

// ---------------------------------------------------------------------------
// Stage 1: one block per (batch, class) channel.
//   async-copy 64KB channel -> LDS (NT), NMS keep-mask on raw values
//   (sigmoid is monotonic), in-place rescore, lazy-tournament top-100.
// ---------------------------------------------------------------------------
extern "C" __global__ __launch_bounds__(NT)
void cn_stage1(const float* __restrict__ heat,
               float* __restrict__ wsScore, int* __restrict__ wsIdx) {
  extern __shared__ char smem1[];
  float* sheat  = (float*)smem1;              // [16384] raw heat, then scores (in place)
  float* redV   = sheat + HWT;                // [8]
  int*   redI   = (int*)(redV + NWAVES);      // [8]
  int*   sBestI = redI + NWAVES;              // [1]

  const int tid = threadIdx.x;
  const int b   = blockIdx.x / NC;
  const int c   = blockIdx.x % NC;
  const float* src = heat + ((size_t)(b * NC + c)) * HWT;

  // ---- async stream channel into LDS: 4096 x B128 (NT), 16 per thread ----
  {
    const uint32_t base = lds_byte_addr(sheat);
    #pragma unroll
    for (int i = 0; i < 16; ++i) {
      const uint32_t off = (uint32_t)((i * NT + tid) * 16);
      async_g2l_b128_nt(base + off, (const char*)src + off);
    }
    wait_async_zero();       // this wave's copies done
  }
  __syncthreads();           // all waves' copies visible

  // ---- Pass 1: NMS keep-mask (keep iff >= all 8 neighbors; borders -inf) ----
  // Thread t owns flat pixels [t*64, t*64+64) = row y=t/2, x in [(t&1)*64, +64).
  uint64_t keepmask = 0;
  const int y  = tid >> 1;
  const int x0 = (tid & 1) * 64;
  for (int j = 0; j < 64; ++j) {
    const int x  = x0 + j;
    const int fi = y * NW_ + x;
    const float v = sheat[fi];
    bool keep = true;
    if (x > 0)        keep &= (v >= sheat[fi - 1]);
    if (x < NW_ - 1)  keep &= (v >= sheat[fi + 1]);
    if (y > 0) {
      keep &= (v >= sheat[fi - NW_]);
      if (x > 0)       keep &= (v >= sheat[fi - NW_ - 1]);
      if (x < NW_ - 1) keep &= (v >= sheat[fi - NW_ + 1]);
    }
    if (y < NH - 1) {
      keep &= (v >= sheat[fi + NW_]);
      if (x > 0)       keep &= (v >= sheat[fi + NW_ - 1]);
      if (x < NW_ - 1) keep &= (v >= sheat[fi + NW_ + 1]);
    }
    keepmask |= (uint64_t)(keep ? 1u : 0u) << j;
  }
  __syncthreads();           // everyone done reading neighbors

  // ---- Pass 2: in-place rescore own segment (no cross-thread reads) ----
  float lmax = -2.0f; int lidx = tid * 64;
  for (int j = 0; j < 64; ++j) {
    const int fi = y * NW_ + x0 + j;
    const float v = sheat[fi];
    const float s = ((keepmask >> j) & 1u) ? (1.0f / (1.0f + __expf(-v))) : 0.0f;
    sheat[fi] = s;
    if (s > lmax) { lmax = s; lidx = fi; }
  }
  __syncthreads();

  // ---- lazy tournament top-100 ----
  float* outS = wsScore + (size_t)b * CANDP + c * KTOP;
  int*   outI = wsIdx   + (size_t)b * CANDP + c * KTOP;
  const int lane = tid & 31, wv = tid >> 5;

  for (int k = 0; k < KTOP; ++k) {
    float v = lmax; int id = lidx;
    #pragma unroll
    for (int off = 16; off > 0; off >>= 1) {
      const float v2 = __shfl_down(v, off, 32);
      const int   i2 = __shfl_down(id, off, 32);
      argmax_pair(v, id, v2, i2);
    }
    if (lane == 0) { redV[wv] = v; redI[wv] = id; }
    __syncthreads();
    if (tid == 0) {
      float bv = redV[0]; int bi = redI[0];
      for (int w = 1; w < NWAVES; ++w) argmax_pair(bv, bi, redV[w], redI[w]);
      outS[k] = bv; outI[k] = bi;
      sBestI[0] = bi;
    }
    __syncthreads();
    const int bi = sBestI[0];
    if ((bi >> 6) == tid) {          // owner removes + rescans its 64 entries
      sheat[bi] = -1.0f;
      float mv = -2.0f; int mi = tid * 64;
      const int base = tid * 64;
      for (int j = 0; j < 64; ++j) {
        const float vv = sheat[base + j];
        if (vv > mv) { mv = vv; mi = base + j; }
      }
      lmax = mv; lidx = mi;
    }
  }
}

// ---------------------------------------------------------------------------
// Stage 2: one block per batch. Global top-100 over 8000 candidates + decode.
// ---------------------------------------------------------------------------
extern "C" __global__ __launch_bounds__(NT)
void cn_stage2(const float* __restrict__ offh, const float* __restrict__ whh,
               const float* __restrict__ wsScore, const int* __restrict__ wsIdx,
               float* __restrict__ out) {
  extern __shared__ char smem2[];
  float* cs     = (float*)smem2;               // [8192] candidate scores
  int*   ci     = (int*)(cs + CANDP);          // [8192] spatial indices
  float* redV   = (float*)(ci + CANDP);        // [8]
  int*   redI   = (int*)(redV + NWAVES);       // [8]
  int*   sBestI = redI + NWAVES;               // [1]
  int*   selJ   = sBestI + 1;                  // [100]
  float* selS   = (float*)(selJ + KTOP);       // [100]

  const int tid = threadIdx.x;
  const int b   = blockIdx.x;
  const float* gS = wsScore + (size_t)b * CANDP;
  const int*   gI = wsIdx   + (size_t)b * CANDP;

  // ---- async load candidates (2 x 32KB), 8 B128 per thread per array ----
  {
    const uint32_t baseS = lds_byte_addr(cs);
    const uint32_t baseI = lds_byte_addr(ci);
    #pragma unroll
    for (int i = 0; i < 8; ++i) {
      const uint32_t off = (uint32_t)((i * NT + tid) * 16);
      async_g2l_b128(baseS + off, (const char*)gS + off);
      async_g2l_b128(baseI + off, (const char*)gI + off);
    }
  }
  // Warm L2 with this batch's offset/wh planes while the DMA runs.
  {
    const char* p0 = (const char*)(offh + (size_t)b * 2 * HWT);
    const char* p1 = (const char*)(whh  + (size_t)b * 2 * HWT);
    #pragma unroll
    for (int i = 0; i < 2; ++i) {
      __builtin_prefetch(p0 + (size_t)(i * NT + tid) * 256, 0, 1);
      __builtin_prefetch(p1 + (size_t)(i * NT + tid) * 256, 0, 1);
    }
  }
  wait_async_zero();
  __syncthreads();
  if (tid < CANDP - NC * KTOP) cs[NC * KTOP + tid] = -1.0f;   // kill pad slots
  __syncthreads();

  // ---- per-thread segment max over 32 entries ----
  float lmax = -2.0f; int lidx = tid * 32;
  for (int j = 0; j < 32; ++j) {
    const float v = cs[tid * 32 + j];
    if (v > lmax) { lmax = v; lidx = tid * 32 + j; }
  }
  __syncthreads();

  const int lane = tid & 31, wv = tid >> 5;
  for (int k = 0; k < KTOP; ++k) {
    float v = lmax; int id = lidx;
    #pragma unroll
    for (int off = 16; off > 0; off >>= 1) {
      const float v2 = __shfl_down(v, off, 32);
      const int   i2 = __shfl_down(id, off, 32);
      argmax_pair(v, id, v2, i2);
    }
    if (lane == 0) { redV[wv] = v; redI[wv] = id; }
    __syncthreads();
    if (tid == 0) {
      float bv = redV[0]; int bi = redI[0];
      for (int w = 1; w < NWAVES; ++w) argmax_pair(bv, bi, redV[w], redI[w]);
      selJ[k] = bi; selS[k] = bv;
      sBestI[0] = bi;
    }
    __syncthreads();
    const int bi = sBestI[0];
    if ((bi >> 5) == tid) {
      cs[bi] = -1.0f;
      float mv = -2.0f; int mi = tid * 32;
      for (int j = 0; j < 32; ++j) {
        const float vv = cs[tid * 32 + j];
        if (vv > mv) { mv = vv; mi = tid * 32 + j; }
      }
      lmax = mv; lidx = mi;
    }
    __syncthreads();
  }

  // ---- decode the 100 winners in parallel ----
  if (tid < KTOP) {
    const int   j   = selJ[tid];
    const float s   = selS[tid];
    const int   cls = j / KTOP;        // candidate list is [class][k]
    const int   idx = ci[j];
    const float xi  = (float)(idx & (NW_ - 1));
    const float yi  = (float)(idx >> 7);
    const float* o0 = offh + (size_t)b * 2 * HWT;
    const float* w0 = whh  + (size_t)b * 2 * HWT;
    const float xs = xi + o0[idx];
    const float ys = yi + o0[HWT + idx];
    const float ww = w0[idx];
    const float hh = w0[HWT + idx];
    const float x1 = fmaxf((xs - 0.5f * ww) * 4.0f, 0.0f);
    const float y1 = fmaxf((ys - 0.5f * hh) * 4.0f, 0.0f);
    const float x2 = fminf((xs + 0.5f * ww) * 4.0f, 511.0f);
    const float y2 = fminf((ys + 0.5f * hh) * 4.0f, 511.0f);
    out[(size_t)b * KTOP + tid]                 = s;           // scores
    out[(size_t)NB * KTOP + b * KTOP + tid]     = (float)cls;  // classes
    float* bb = out + (size_t)2 * NB * KTOP + ((size_t)b * KTOP + tid) * 4;
    bb[0] = x1; bb[1] = y1; bb[2] = x2; bb[3] = y2;            // bboxes
  }
}

extern "C" void kernel_launch(void* const* d_in, const int* in_sizes, int n_in,
                              void* d_out, int out_size, void* d_ws, size_t ws_size,
                              hipStream_t stream) {
  const float* heat = (const float*)d_in[0];   // [16,80,128,128]
  const float* offh = (const float*)d_in[1];   // [16,2,128,128]
  const float* whh  = (const float*)d_in[2];   // [16,2,128,128]
  float* out = (float*)d_out;                  // scores|classes|bboxes = 9600 floats

  // Workspace: per-batch padded candidate lists (pad keeps B128 DMA in-bounds).
  float* wsScore = (float*)d_ws;                                   // 16*8192 f32
  int*   wsIdx   = (int*)((char*)d_ws +
                          (size_t)NB * CANDP * sizeof(float));     // 16*8192 i32

  const size_t shm1 = (size_t)HWT * sizeof(float)                  // sheat (in-place)
                    + (2 * NWAVES + 1) * sizeof(float) + 64;       // ~64.1 KB
  cn_stage1<<<dim3(NB * NC), dim3(NT), shm1, stream>>>(heat, wsScore, wsIdx);

  const size_t shm2 = (size_t)(2 * CANDP) * sizeof(float)
                    + (2 * NWAVES + 1 + 2 * KTOP) * sizeof(float) + 64;  // ~65 KB
  cn_stage2<<<dim3(NB), dim3(NT), shm2, stream>>>(offh, whh, wsScore, wsIdx, out);
}